// GaussianModel_90890097918439
// MI455X (gfx1250) — compile-verified
//
#include <hip/hip_runtime.h>
#include <math.h>

typedef __attribute__((ext_vector_type(2))) float v2f;
typedef __attribute__((ext_vector_type(8))) float v8f;

#define IMG_W 128
#define IMG_H 128

// ---------------------------------------------------------------------------
// Kernel 1: per-gaussian preprocessing (projection, conic, SH color, opacity)
// ---------------------------------------------------------------------------
__global__ void gs_preprocess(const float* __restrict__ means,
                              const float* __restrict__ scales,
                              const float* __restrict__ quats,
                              const float* __restrict__ opacities,
                              const float* __restrict__ sh0,
                              const float* __restrict__ shN,
                              const float* __restrict__ viewm,
                              const float* __restrict__ Kc,
                              const int*   __restrict__ shdeg_p,
                              float* __restrict__ tz_out,
                              float* __restrict__ rec,
                              int N)
{
    int g = blockIdx.x * blockDim.x + threadIdx.x;
    if (g >= N) return;

    float R[3][3], t[3];
    #pragma unroll
    for (int i = 0; i < 3; i++) {
        #pragma unroll
        for (int j = 0; j < 3; j++) R[i][j] = viewm[i*4+j];
        t[i] = viewm[i*4+3];
    }
    float fx = Kc[0], fy = Kc[4], cx = Kc[2], cy = Kc[5];

    float m0 = means[g*3+0], m1 = means[g*3+1], m2 = means[g*3+2];
    float px  = R[0][0]*m0 + R[0][1]*m1 + R[0][2]*m2 + t[0];
    float py_ = R[1][0]*m0 + R[1][1]*m1 + R[1][2]*m2 + t[1];
    float pz  = R[2][0]*m0 + R[2][1]*m1 + R[2][2]*m2 + t[2];
    float iz  = 1.0f / pz;
    float mx  = fx*px*iz + cx;
    float my  = fy*py_*iz + cy;

    // quaternion -> rotation
    float qw = quats[g*4+0], qx = quats[g*4+1], qy = quats[g*4+2], qz = quats[g*4+3];
    float qn = rsqrtf(qw*qw + qx*qx + qy*qy + qz*qz);
    qw *= qn; qx *= qn; qy *= qn; qz *= qn;
    float Rg[3][3];
    Rg[0][0] = 1.f-2.f*(qy*qy+qz*qz); Rg[0][1] = 2.f*(qx*qy-qw*qz); Rg[0][2] = 2.f*(qx*qz+qw*qy);
    Rg[1][0] = 2.f*(qx*qy+qw*qz); Rg[1][1] = 1.f-2.f*(qx*qx+qz*qz); Rg[1][2] = 2.f*(qy*qz-qw*qx);
    Rg[2][0] = 2.f*(qx*qz-qw*qy); Rg[2][1] = 2.f*(qy*qz+qw*qx); Rg[2][2] = 1.f-2.f*(qx*qx+qy*qy);

    float s0 = __expf(scales[g*3+0]), s1 = __expf(scales[g*3+1]), s2 = __expf(scales[g*3+2]);
    float M[3][3];
    #pragma unroll
    for (int i = 0; i < 3; i++) { M[i][0]=Rg[i][0]*s0; M[i][1]=Rg[i][1]*s1; M[i][2]=Rg[i][2]*s2; }
    float C3[3][3];
    #pragma unroll
    for (int i = 0; i < 3; i++)
        #pragma unroll
        for (int j = 0; j < 3; j++)
            C3[i][j] = M[i][0]*M[j][0] + M[i][1]*M[j][1] + M[i][2]*M[j][2];
    float RC[3][3], CC[3][3];
    #pragma unroll
    for (int i = 0; i < 3; i++)
        #pragma unroll
        for (int j = 0; j < 3; j++)
            RC[i][j] = R[i][0]*C3[0][j] + R[i][1]*C3[1][j] + R[i][2]*C3[2][j];
    #pragma unroll
    for (int i = 0; i < 3; i++)
        #pragma unroll
        for (int j = 0; j < 3; j++)
            CC[i][j] = RC[i][0]*R[j][0] + RC[i][1]*R[j][1] + RC[i][2]*R[j][2];

    float J[2][3] = {{fx*iz, 0.f, -fx*px*iz*iz}, {0.f, fy*iz, -fy*py_*iz*iz}};
    float JC[2][3];
    #pragma unroll
    for (int i = 0; i < 2; i++)
        #pragma unroll
        for (int j = 0; j < 3; j++)
            JC[i][j] = J[i][0]*CC[0][j] + J[i][1]*CC[1][j] + J[i][2]*CC[2][j];
    float a = JC[0][0]*J[0][0] + JC[0][1]*J[0][1] + JC[0][2]*J[0][2] + 0.3f;
    float b = JC[0][0]*J[1][0] + JC[0][1]*J[1][1] + JC[0][2]*J[1][2];
    float c = JC[1][0]*J[1][0] + JC[1][1]*J[1][1] + JC[1][2]*J[1][2] + 0.3f;
    float det  = a*c - b*b;
    float idet = 1.0f / det;
    float ca = c*idet, cb = -b*idet, cc = a*idet;

    // view dir + SH color
    float cpx = -(R[0][0]*t[0] + R[1][0]*t[1] + R[2][0]*t[2]);
    float cpy = -(R[0][1]*t[0] + R[1][1]*t[1] + R[2][1]*t[2]);
    float cpz = -(R[0][2]*t[0] + R[1][2]*t[1] + R[2][2]*t[2]);
    float dx = m0-cpx, dy = m1-cpy, dz = m2-cpz;
    float dn = rsqrtf(dx*dx + dy*dy + dz*dz);
    dx *= dn; dy *= dn; dz *= dn;
    float xx = dx*dx, yy = dy*dy, zz = dz*dz;
    float basis[16];
    basis[0]  = 0.28209479177387814f;
    basis[1]  = -0.4886025119029199f*dy;
    basis[2]  = 0.4886025119029199f*dz;
    basis[3]  = -0.4886025119029199f*dx;
    basis[4]  = 1.0925484305920792f*dx*dy;
    basis[5]  = -1.0925484305920792f*dy*dz;
    basis[6]  = 0.31539156525252005f*(2.f*zz-xx-yy);
    basis[7]  = -1.0925484305920792f*dx*dz;
    basis[8]  = 0.5462742152960396f*(xx-yy);
    basis[9]  = -0.5900435899266435f*dy*(3.f*xx-yy);
    basis[10] = 2.890611442640554f*dx*dy*dz;
    basis[11] = -0.4570457994644658f*dy*(4.f*zz-xx-yy);
    basis[12] = 0.3731763325901154f*dz*(2.f*zz-3.f*xx-3.f*yy);
    basis[13] = -0.4570457994644658f*dx*(4.f*zz-xx-yy);
    basis[14] = 1.445305721320277f*dz*(xx-yy);
    basis[15] = -0.5900435899266435f*dx*(xx-yy);

    int deg = shdeg_p[0];
    int nsh = (deg+1)*(deg+1);
    if (nsh > 16) nsh = 16;
    float col[3];
    #pragma unroll
    for (int ch = 0; ch < 3; ++ch) {
        float acc = basis[0] * sh0[g*3+ch];
        for (int k = 1; k < nsh; k++) acc += basis[k] * shN[(g*15 + (k-1))*3 + ch];
        col[ch] = fmaxf(acc + 0.5f, 0.0f);
    }
    float op = 1.0f / (1.0f + __expf(-opacities[g]));

    bool valid = (pz > 0.01f) && (det > 1e-12f);
    if (!valid) { op = 0.0f; ca = 0.0f; cb = 0.0f; cc = 0.0f; mx = 0.0f; my = 0.0f; }

    tz_out[g] = pz;
    float* r_ = rec + g*12;
    r_[0]=mx; r_[1]=my; r_[2]=ca; r_[3]=cb; r_[4]=cc; r_[5]=op;
    r_[6]=col[0]; r_[7]=col[1]; r_[8]=col[2]; r_[9]=0.f; r_[10]=0.f; r_[11]=0.f;
}

// ---------------------------------------------------------------------------
// Kernel 2: stable rank sort by depth (matches jnp.argsort stability) +
// scatter into WMMA A-matrix coefficient rows.
//   sigma(p,g) = dot(coef[g][0..5], [1, gx, gy, gx^2, gx*gy, gy^2])
// ---------------------------------------------------------------------------
__global__ void gs_sort_scatter(const float* __restrict__ tz,
                                const float* __restrict__ rec,
                                float* __restrict__ coef,
                                float* __restrict__ scal,
                                int N)
{
    int g = blockIdx.x * blockDim.x + threadIdx.x;
    if (g >= N) return;
    float myz = tz[g];
    int rank = 0;
    for (int j = 0; j < N; j++) {
        float zj = tz[j];
        rank += (zj < myz) || (zj == myz && j < g);
    }
    const float* r_ = rec + g*12;
    float mx = r_[0], my = r_[1], ca = r_[2], cb = r_[3], cc = r_[4];
    float* cf = coef + rank*8;
    cf[0] = 0.5f*ca*mx*mx + 0.5f*cc*my*my + cb*mx*my;
    cf[1] = -(ca*mx + cb*my);
    cf[2] = -(cc*my + cb*mx);
    cf[3] = 0.5f*ca;
    cf[4] = cb;
    cf[5] = 0.5f*cc;
    cf[6] = 0.f;
    cf[7] = 0.f;
    float* sc = scal + rank*4;
    sc[0] = r_[5]; sc[1] = r_[6]; sc[2] = r_[7]; sc[3] = r_[8];
}

// ---------------------------------------------------------------------------
// Kernel 3: fused splat/blend. One wave per 16 pixels; gaussians in chunks of
// 16. Sigma tile (16 gauss x 16 pix) computed by two V_WMMA_F32_16X16X4_F32.
// D layout: lane L -> pixel N=L&15, gaussian rows M = (L>=16?8:0)+r (depth
// order), so the transmittance scan runs in-lane over 8 gaussians, with one
// shfl_xor(16) per chunk to stitch the halves (blend is affine in chunk-start
// T). XDL WMMA co-executes with the VALU exp/blend math.
// ---------------------------------------------------------------------------
__global__ void __launch_bounds__(256)
gs_render(const float* __restrict__ coef,
          const float* __restrict__ scal,
          const float* __restrict__ bg,
          float* __restrict__ out,
          int N)
{
    int tid   = blockIdx.x * blockDim.x + threadIdx.x;
    int lane  = threadIdx.x & 31;
    int gwave = tid >> 5;
    int pix   = gwave*16 + (lane & 15);
    float gx = (float)(pix % IMG_W) + 0.5f;
    float gy = (float)(pix / IMG_W) + 0.5f;

    // A(16x4 f32): lanes 0-15 supply K=0,1; lanes 16-31 K=2,3 (same rows).
    // B(4x16 f32): mirrored. base selects which half of the 8-dim feature
    // vector this lane feeds per WMMA.
    int base = (lane >= 16) ? 2 : 0;
    float f[8] = {1.f, gx, gy, gx*gx, gx*gy, gy*gy, 0.f, 0.f};
    v2f B1; B1.x = f[base];     B1.y = f[base+1];
    v2f B2; B2.x = f[4+base];   B2.y = f[4+base+1];

    int mrow = (lane >= 16) ? 8 : 0;   // this lane's gaussian rows in D
    int ga   = (lane & 15);            // gaussian row this lane loads for A

    float T = 1.0f;
    float rgb0 = 0.f, rgb1 = 0.f, rgb2 = 0.f;
    int chunks = N >> 4;

    for (int c0 = 0; c0 < chunks; ++c0) {
        int gbase = c0 << 4;
        const float* cfp = coef + (size_t)(gbase + ga)*8 + base;
        v2f A1; A1.x = cfp[0]; A1.y = cfp[1];
        v2f A2; A2.x = cfp[4]; A2.y = cfp[5];

        v8f acc = {0.f,0.f,0.f,0.f,0.f,0.f,0.f,0.f};
        acc = __builtin_amdgcn_wmma_f32_16x16x4_f32(false, A1, false, B1,
                                                    (short)0, acc, false, false);
        acc = __builtin_amdgcn_wmma_f32_16x16x4_f32(false, A2, false, B2,
                                                    (short)0, acc, false, false);

        // in-lane blend over this half's 8 depth-ordered gaussians,
        // relative to T=1 at chunk start
        float run = 1.0f;
        float rp0 = 0.f, rp1 = 0.f, rp2 = 0.f;
        #pragma unroll
        for (int r = 0; r < 8; r++) {
            const float* sp = scal + (size_t)(gbase + mrow + r)*4;
            float op = sp[0], cr = sp[1], cg = sp[2], cbl = sp[3];
            float sigma = acc[r];
            float a = op * __expf(-sigma);
            bool keep = (sigma >= 0.0f) && (a > (1.0f/255.0f));
            a = keep ? fminf(a, 0.999f) : 0.0f;
            float w = a * run;
            rp0 += w*cr; rp1 += w*cg; rp2 += w*cbl;
            run *= (1.0f - a);
        }
        // stitch halves: high half's start-T includes low half's product
        float other  = __shfl_xor(run, 16, 32);
        float tscale = T * ((lane >= 16) ? other : 1.0f);
        rgb0 += tscale*rp0; rgb1 += tscale*rp1; rgb2 += tscale*rp2;
        T *= run * other;
    }

    // merge partial rgb between lane p and p+16
    float r0 = rgb0 + __shfl_xor(rgb0, 16, 32);
    float r1 = rgb1 + __shfl_xor(rgb1, 16, 32);
    float r2 = rgb2 + __shfl_xor(rgb2, 16, 32);

    if (lane < 16) {
        out[pix*3+0] = r0 + T*bg[0];
        out[pix*3+1] = r1 + T*bg[1];
        out[pix*3+2] = r2 + T*bg[2];
        out[IMG_W*IMG_H*3 + pix] = 1.0f - T;
    }
}

// ---------------------------------------------------------------------------
extern "C" void kernel_launch(void* const* d_in, const int* in_sizes, int n_in,
                              void* d_out, int out_size, void* d_ws, size_t ws_size,
                              hipStream_t stream)
{
    const float* means  = (const float*)d_in[0];
    const float* scales = (const float*)d_in[1];
    const float* quats  = (const float*)d_in[2];
    const float* opac   = (const float*)d_in[3];
    const float* sh0    = (const float*)d_in[4];
    const float* shN    = (const float*)d_in[5];
    const float* viewm  = (const float*)d_in[6];
    const float* Kc     = (const float*)d_in[7];
    const float* bg     = (const float*)d_in[8];
    const int*   shdeg  = (const int*)d_in[11];

    int N = in_sizes[3];            // N_GAUSS (opacities is N-long)

    float* ws   = (float*)d_ws;
    float* tz   = ws;               // N floats
    float* rec  = ws + N;           // N*12 floats
    float* coef = ws + (size_t)N*13;// N*8 floats (32B-aligned rows)
    float* scal = ws + (size_t)N*21;// N*4 floats

    gs_preprocess<<<(N+255)/256, 256, 0, stream>>>(
        means, scales, quats, opac, sh0, shN, viewm, Kc, shdeg, tz, rec, N);
    gs_sort_scatter<<<(N+255)/256, 256, 0, stream>>>(tz, rec, coef, scal, N);

    int waves   = (IMG_W*IMG_H) / 16;       // 1024 waves, 16 pixels each
    int threads = waves * 32;               // 32768
    gs_render<<<threads/256, 256, 0, stream>>>(coef, scal, bg, (float*)d_out, N);
}